// GCN_82377472737539
// MI455X (gfx1250) — compile-verified
//
#include <hip/hip_runtime.h>
#include <math.h>

typedef __attribute__((ext_vector_type(2))) float v2f;
typedef __attribute__((ext_vector_type(8))) float v8f;

#define N_NODES   100000
#define N_EDGES   1600000
#define N_FEAT    128
#define HIDDEN    128
#define N_CLASSES 10
#define N_GRAPHS  100
#define BN_EPS    1e-5f

// ---------------------------------------------------------------- utilities
__global__ void gcn_zero_kernel(float* __restrict__ p, int n) {
    int i = blockIdx.x * blockDim.x + threadIdx.x;
    if (i < n) p[i] = 0.0f;
}

// deg[dst] += 1 per edge
__global__ void gcn_degree_kernel(const int* __restrict__ dst, float* __restrict__ deg, int nE) {
    int e = blockIdx.x * blockDim.x + threadIdx.x;
    if (e < nE) atomicAdd(&deg[dst[e]], 1.0f);
}

// dis = rsqrt(deg + 1)  (self loop)
__global__ void gcn_deg_finalize_kernel(float* __restrict__ dis, int n) {
    int i = blockIdx.x * blockDim.x + threadIdx.x;
    if (i < n) dis[i] = rsqrtf(dis[i] + 1.0f);
}

// ----------------------------------------------------- WMMA fp32 GEMM  h=A@B
// A: [M x 128] row-major, B: [128 x 128] row-major, C: [M x 128] row-major.
// Block = 256 threads = 8 waves. B (64 KB) is staged once into LDS, then the
// block grid-strides over 16-row tiles (persistent blocks). Wave w computes
// the 16x16 tile at cols [16*w, +16), accumulating K=128 in steps of 4 with
// V_WMMA_F32_16X16X4_F32.
__global__ void gcn_gemm128_wmma_kernel(const float* __restrict__ A,
                                        const float* __restrict__ B,
                                        float* __restrict__ C, int M) {
    __shared__ float Bs[128 * 128];      // 64 KB of the 320 KB/WGP LDS

    const int tid = threadIdx.x;
    // cooperative B load: 16384 floats / 256 threads = 16 float4 each
    for (int i = tid * 4; i < 128 * 128; i += 256 * 4) {
        *(float4*)&Bs[i] = *(const float4*)&B[i];
    }
    __syncthreads();

    const int wave = tid >> 5;           // 0..7
    const int lane = tid & 31;
    const int half = lane >> 4;          // 0 or 1
    const int lr   = lane & 15;          // 0..15
    const int col0 = wave * 16;
    const float* __restrict__ bcol = Bs + col0 + lr;

    const int nTiles = M >> 4;
    for (int tile = blockIdx.x; tile < nTiles; tile += gridDim.x) {
        const int row0 = tile * 16;
        const float* __restrict__ arow = A + (size_t)(row0 + lr) * 128;

        // prefetch next tile's A rows (global_prefetch_b8)
        int nt = tile + gridDim.x;
        if (nt < nTiles) {
            __builtin_prefetch(A + (size_t)(nt * 16 + lr) * 128, 0, 0);
        }

        v8f c = {};
#pragma unroll
        for (int k0 = 0; k0 < 128; k0 += 4) {
            const int ka = k0 + 2 * half;
            v2f a, b;
            // A 16x4: lane lr holds row lr, elements (ka, ka+1) -> b64 load
            a.x = arow[ka];
            a.y = arow[ka + 1];
            // B 4x16 from LDS: lane lr holds col lr, rows (ka, ka+1)
            b.x = bcol[ka * 128];
            b.y = bcol[(ka + 1) * 128];
            c = __builtin_amdgcn_wmma_f32_16x16x4_f32(false, a, false, b,
                                                      (short)0, c, false, false);
        }
        // D layout: VGPR v holds element [v + 8*half][lr]
#pragma unroll
        for (int v = 0; v < 8; ++v) {
            C[(size_t)(row0 + v + 8 * half) * 128 + col0 + lr] = c[v];
        }
    }
}

// ------------------------------------------- agg = h * dis^2 + bias (self loop)
__global__ void gcn_selfinit_kernel(const float* __restrict__ h,
                                    const float* __restrict__ dis,
                                    const float* __restrict__ bias,
                                    float* __restrict__ agg, int n) {
    int idx = blockIdx.x * blockDim.x + threadIdx.x;
    if (idx >= n * HIDDEN) return;
    int i = idx >> 7;
    int f = idx & 127;
    float d = dis[i];
    agg[idx] = h[idx] * d * d + bias[f];
}

// ------------------------------------- edge scatter: agg[dst] += h[src]*norm
// one wave per edge; each lane handles a float4 (32 lanes * 4 = 128 features)
__global__ void gcn_edge_scatter_kernel(const int* __restrict__ src,
                                        const int* __restrict__ dst,
                                        const float* __restrict__ dis,
                                        const float* __restrict__ h,
                                        float* __restrict__ agg, int nE) {
    int gw     = (int)((blockIdx.x * blockDim.x + threadIdx.x) >> 5);
    int lane   = threadIdx.x & 31;
    int stride = (int)((gridDim.x * blockDim.x) >> 5);
    for (int e = gw; e < nE; e += stride) {
        int s = src[e];
        int d = dst[e];
        float norm = dis[s] * dis[d];
        const float4* hp = (const float4*)(h + (size_t)s * 128);
        float4 v = hp[lane];
        float* ap = agg + (size_t)d * 128 + lane * 4;
        atomicAdd(ap + 0, v.x * norm);
        atomicAdd(ap + 1, v.y * norm);
        atomicAdd(ap + 2, v.z * norm);
        atomicAdd(ap + 3, v.w * norm);
    }
}

// --------------------------------------- per-column sum / sumsq for BatchNorm
__global__ void gcn_stats_kernel(const float* __restrict__ x,
                                 float* __restrict__ st, int n) {
    int f = threadIdx.x;                 // blockDim.x == 128
    float s = 0.0f, s2 = 0.0f;
    for (int r = blockIdx.x; r < n; r += gridDim.x) {
        float v = x[(size_t)r * 128 + f];
        s += v;
        s2 += v * v;
    }
    atomicAdd(&st[f], s);
    atomicAdd(&st[128 + f], s2);
}

// --------------------------------------------------- BN (train mode) + ReLU
__global__ void gcn_bnrelu_kernel(float* __restrict__ x,
                                  const float* __restrict__ st,
                                  const float* __restrict__ gamma,
                                  const float* __restrict__ beta, int n) {
    int idx = blockIdx.x * blockDim.x + threadIdx.x;
    if (idx >= n * HIDDEN) return;
    int f = idx & 127;
    const float invn = 1.0f / (float)n;
    float mean = st[f] * invn;
    float var  = st[128 + f] * invn - mean * mean;
    float inv  = rsqrtf(var + BN_EPS);
    float v = (x[idx] - mean) * inv * gamma[f] + beta[f];
    x[idx] = v > 0.0f ? v : 0.0f;
}

// ------------------------------------------------------ mean-pool accumulate
__global__ void gcn_pool_kernel(const float* __restrict__ x,
                                const int* __restrict__ batch,
                                float* __restrict__ ps,
                                float* __restrict__ cnt, int n) {
    int gw     = (int)((blockIdx.x * blockDim.x + threadIdx.x) >> 5);
    int lane   = threadIdx.x & 31;
    int stride = (int)((gridDim.x * blockDim.x) >> 5);
    for (int i = gw; i < n; i += stride) {
        int g = batch[i];
        const float4* xp = (const float4*)(x + (size_t)i * 128);
        float4 v = xp[lane];
        float* pp = ps + (size_t)g * 128 + lane * 4;
        atomicAdd(pp + 0, v.x);
        atomicAdd(pp + 1, v.y);
        atomicAdd(pp + 2, v.z);
        atomicAdd(pp + 3, v.w);
        if (lane == 0) atomicAdd(&cnt[g], 1.0f);
    }
}

// ------------------- MLP head + log_softmax: one block (128 thr) per graph
__global__ void gcn_head_kernel(const float* __restrict__ ps,
                                const float* __restrict__ cnt,
                                const float* __restrict__ w1,
                                const float* __restrict__ b1,
                                const float* __restrict__ w2,
                                const float* __restrict__ b2,
                                float* __restrict__ out) {
    __shared__ float pooled[128];
    __shared__ float h1[128];
    __shared__ float logits[N_CLASSES];
    int g = blockIdx.x;
    int t = threadIdx.x;                 // 0..127
    float c = cnt[g];
    c = c > 1.0f ? c : 1.0f;
    pooled[t] = ps[(size_t)g * 128 + t] / c;
    __syncthreads();
    float acc = b1[t];
    for (int k = 0; k < 128; ++k) acc += pooled[k] * w1[(size_t)k * 128 + t];
    h1[t] = acc > 0.0f ? acc : 0.0f;
    __syncthreads();
    if (t < N_CLASSES) {
        float a = b2[t];
        for (int k = 0; k < 128; ++k) a += h1[k] * w2[(size_t)k * N_CLASSES + t];
        logits[t] = a;
    }
    __syncthreads();
    if (t == 0) {
        float mx = logits[0];
        for (int i = 1; i < N_CLASSES; ++i) mx = fmaxf(mx, logits[i]);
        float se = 0.0f;
        for (int i = 0; i < N_CLASSES; ++i) se += expf(logits[i] - mx);
        float lse = mx + logf(se);
        for (int i = 0; i < N_CLASSES; ++i) out[(size_t)g * N_CLASSES + i] = logits[i] - lse;
    }
}

// ---------------------------------------------------------------------------
extern "C" void kernel_launch(void* const* d_in, const int* in_sizes, int n_in,
                              void* d_out, int out_size, void* d_ws, size_t ws_size,
                              hipStream_t stream) {
    (void)in_sizes; (void)n_in; (void)out_size; (void)ws_size;

    const float* x_in  = (const float*)d_in[0];
    const int*   ei    = (const int*)d_in[1];
    const int*   batch = (const int*)d_in[2];
    const int*   src   = ei;
    const int*   dst   = ei + N_EDGES;

    const float* W[3]     = {(const float*)d_in[3],  (const float*)d_in[7],  (const float*)d_in[11]};
    const float* bia[3]   = {(const float*)d_in[4],  (const float*)d_in[8],  (const float*)d_in[12]};
    const float* gamma[3] = {(const float*)d_in[5],  (const float*)d_in[9],  (const float*)d_in[13]};
    const float* beta[3]  = {(const float*)d_in[6],  (const float*)d_in[10], (const float*)d_in[14]};
    const float* lin1_w = (const float*)d_in[15];
    const float* lin1_b = (const float*)d_in[16];
    const float* lin2_w = (const float*)d_in[17];
    const float* lin2_b = (const float*)d_in[18];
    float* out = (float*)d_out;

    // ---- workspace carve (256B aligned chunks) ----
    size_t off = 0;
    char* base = (char*)d_ws;
    auto carve = [&](size_t nfloats) -> float* {
        float* p = (float*)(base + off);
        off += ((nfloats * sizeof(float) + 255) / 256) * 256;
        return p;
    };
    float* dis = carve(N_NODES);            // deg -> rsqrt(deg+1)
    float* h   = carve((size_t)N_NODES * HIDDEN);
    float* agg = carve((size_t)N_NODES * HIDDEN);
    float* st  = carve(256);                // col sum / sumsq
    float* ps  = carve((size_t)N_GRAPHS * HIDDEN);
    float* cg  = carve(N_GRAPHS);

    const int elemN = N_NODES * HIDDEN;

    // ---- degrees ----
    gcn_zero_kernel<<<(N_NODES + 255) / 256, 256, 0, stream>>>(dis, N_NODES);
    gcn_degree_kernel<<<(N_EDGES + 255) / 256, 256, 0, stream>>>(dst, dis, N_EDGES);
    gcn_deg_finalize_kernel<<<(N_NODES + 255) / 256, 256, 0, stream>>>(dis, N_NODES);

    // ---- 3 GCN layers ----
    const int gemmBlocks    = 1250;                         // 5 row tiles per block
    const int scatterBlocks = (N_EDGES * 32 + 255) / 256;   // 1 wave per edge
    for (int l = 0; l < 3; ++l) {
        const float* xin = (l == 0) ? x_in : agg;           // BN output lives in agg
        gcn_gemm128_wmma_kernel<<<gemmBlocks, 256, 0, stream>>>(xin, W[l], h, N_NODES);
        gcn_selfinit_kernel<<<(elemN + 255) / 256, 256, 0, stream>>>(h, dis, bia[l], agg, N_NODES);
        gcn_edge_scatter_kernel<<<scatterBlocks, 256, 0, stream>>>(src, dst, dis, h, agg, N_EDGES);
        gcn_zero_kernel<<<1, 256, 0, stream>>>(st, 256);
        gcn_stats_kernel<<<512, 128, 0, stream>>>(agg, st, N_NODES);
        gcn_bnrelu_kernel<<<(elemN + 255) / 256, 256, 0, stream>>>(agg, st, gamma[l], beta[l], N_NODES);
    }

    // ---- global mean pool + head ----
    gcn_zero_kernel<<<(N_GRAPHS * HIDDEN + 255) / 256, 256, 0, stream>>>(ps, N_GRAPHS * HIDDEN);
    gcn_zero_kernel<<<1, 128, 0, stream>>>(cg, N_GRAPHS);
    gcn_pool_kernel<<<(N_NODES * 32 + 255) / 256, 256, 0, stream>>>(agg, batch, ps, cg, N_NODES);
    gcn_head_kernel<<<N_GRAPHS, 128, 0, stream>>>(ps, cg, lin1_w, lin1_b, lin2_w, lin2_b, out);
}